// VisualBERT_58007828299916
// MI455X (gfx1250) — compile-verified
//
#include <hip/hip_runtime.h>
#include <hip/hip_bf16.h>
#include <stdint.h>

#define NUM_CLASSES 80
#define NB 8
#define NN 512
#define IMG_W 1333.0f
#define IMG_H 800.0f
#define NMS_THRESH 0.5f
#define SCORE_THRESH 0.5f

typedef __attribute__((ext_vector_type(2))) float v2f;
typedef __attribute__((ext_vector_type(8))) float v8f;

// ---------------- Kernel 1: softmax -> transposed scores, zero max accumulator ---------
__global__ void prep_kernel(const float* __restrict__ logits,
                            float* __restrict__ wsmax,
                            float* __restrict__ scores_t) {
  int row = blockIdx.x * blockDim.x + threadIdx.x;   // row = b*512 + n
  if (row >= NB * NN) return;
  wsmax[row] = 0.0f;                                  // deterministic re-init every call
  const float* lp = logits + (size_t)row * (NUM_CLASSES + 1);
  float mx = -1e30f;
  for (int k = 0; k <= NUM_CLASSES; ++k) mx = fmaxf(mx, lp[k]);
  float sum = 0.0f;
  for (int k = 0; k <= NUM_CLASSES; ++k) sum += expf(lp[k] - mx);
  float inv = 1.0f / sum;
  int b = row >> 9, n = row & (NN - 1);
  float* op = scores_t + (size_t)b * NUM_CLASSES * NN + n;
  for (int c = 0; c < NUM_CLASSES; ++c)
    op[(size_t)c * NN] = expf(lp[c + 1] - mx) * inv;  // probs[..., 1:] transposed to [b][c][n]
}

// ---------------- Kernel 2: per-(b,c) NMS, one workgroup each -------------------------
__global__ __launch_bounds__(512)
void nms_kernel(const float* __restrict__ boxes,
                const float* __restrict__ scale,
                const float* __restrict__ scores_t,
                float* __restrict__ wsmax) {
  __shared__ float    s_val[NN];
  __shared__ int      s_idx[NN];
  __shared__ float4   s_box[NN];         // (x1,y1,x2,y2) packed -> one ds_load_b128 per box
  __shared__ float    s_area[NN];
  __shared__ unsigned s_mask[NN * 16];   // 512 x 512-bit suppression matrix
  __shared__ unsigned s_keep[16];

  const int t    = threadIdx.x;
  const int bc   = blockIdx.x;
  const int b    = bc / NUM_CLASSES;
  const int c    = bc % NUM_CLASSES;
  const int lane = t & 31;
  const int wave = t >> 5;

  // ---- stage scores into LDS via CDNA5 async Global->LDS path (ASYNCcnt) ----
  {
    unsigned long long ga = (unsigned long long)(uintptr_t)(scores_t + (size_t)bc * NN + t);
    unsigned lds = (unsigned)(uintptr_t)(&s_val[t]);   // low 32 bits of flat LDS addr = LDS offset
    asm volatile("global_load_async_to_lds_b32 %0, %1, off"
                 :: "v"(lds), "v"(ga) : "memory");
  }
  // Prefetch the box rows: the address set {boxes[b,n,c,:]} is permutation-invariant.
  __builtin_prefetch(boxes + (((size_t)(b * NN + t)) * NUM_CLASSES + c) * 4, 0, 1);
#pragma unroll
  for (int k = 0; k < 16; ++k) s_mask[t * 16 + k] = 0u;
  asm volatile("s_wait_asynccnt 0x0" ::: "memory");
  __syncthreads();

  // ---- bitonic sort (descending, payload = index), hybrid register/LDS ----
  // Stages with partner distance j<32 run in-wave via ds_bpermute (no barriers);
  // only j>=32 stages (10 of 45) touch LDS arrays with barriers.
  float rv = s_val[t];
  int   ri = t;
  for (int k = 2; k <= NN; k <<= 1) {
    int j = k >> 1;
    for (; j >= 32; j >>= 1) {           // cross-wave stages via LDS
      s_val[t] = rv; s_idx[t] = ri;
      __syncthreads();
      int p = t ^ j;
      float pv = s_val[p];
      int   pi = s_idx[p];
      bool dir     = ((t & k) == 0);     // descending run
      bool lower   = ((t & j) == 0);
      bool keepMax = (dir == lower);
      bool take    = keepMax ? (pv > rv) : (pv < rv);
      if (take) { rv = pv; ri = pi; }
      __syncthreads();
    }
    for (; j > 0; j >>= 1) {             // in-wave stages, barrier-free
      int addr = ((lane ^ j) << 2);
      float pv = __int_as_float(__builtin_amdgcn_ds_bpermute(addr, __float_as_int(rv)));
      int   pi = __builtin_amdgcn_ds_bpermute(addr, ri);
      bool dir     = ((t & k) == 0);
      bool lower   = ((t & j) == 0);
      bool keepMax = (dir == lower);
      bool take    = keepMax ? (pv > rv) : (pv < rv);
      if (take) { rv = pv; ri = pi; }
    }
  }
  s_val[t] = rv; s_idx[t] = ri;
  __syncthreads();

  // ---- gather boxes in sorted order, scale + clip, compute areas ----
  {
    int n = s_idx[t];
    const float4 bx = *(const float4*)(boxes + (((size_t)(b * NN + n)) * NUM_CLASSES + c) * 4);
    float sx = scale[b * 2 + 0], sy = scale[b * 2 + 1];
    float x1 = fminf(fmaxf(bx.x * sx, 0.0f), IMG_W);
    float y1 = fminf(fmaxf(bx.y * sy, 0.0f), IMG_H);
    float x2 = fminf(fmaxf(bx.z * sx, 0.0f), IMG_W);
    float y2 = fminf(fmaxf(bx.w * sy, 0.0f), IMG_H);
    s_box[t] = make_float4(x1, y1, x2, y2);
    s_area[t] = fmaxf(x2 - x1, 0.0f) * fmaxf(y2 - y1, 0.0f);
  }
  __syncthreads();

  // ---- suppression bit-matrix: 528 upper-triangle 16x16 tiles over 16 waves ----
  // WMMA computes the pairwise area sum C[m][n] = area_i[m] + area_j[n] (union term).
  const int  ml = lane & 15;
  const bool hiHalf = (lane >= 16);
  for (int tile = wave; tile < 528; tile += 16) {
    int u = tile, ti = 0, rem = 32;
    while (u >= rem) { u -= rem; ++ti; --rem; }   // wave-uniform triangular decode
    int tj = ti + u;
    int i0 = ti << 4, j0 = tj << 4;

    // A[m][k] = area_i[m] for K={0,1}, 0.5 for K={2,3};  B[k][n] = 0.5 for K={0,1}, area_j[n] for K={2,3}
    float av = hiHalf ? 0.5f : s_area[i0 + ml];
    float bv = hiHalf ? s_area[j0 + ml] : 0.5f;
    v2f A;  A[0] = av;  A[1] = av;
    v2f Bm; Bm[0] = bv; Bm[1] = bv;
    v8f C = {};
    C = __builtin_amdgcn_wmma_f32_16x16x4_f32(false, A, false, Bm, (short)0, C, false, false);

    int gj = j0 + ml;                     // column index for this lane (N = lane & 15)
    float4 bj = s_box[gj];
#pragma unroll
    for (int r = 0; r < 8; ++r) {
      int gi = i0 + r + (hiHalf ? 8 : 0); // row index: M = r + 8*(lane>=16)
      float4 bi = s_box[gi];              // uniform per half-wave -> LDS broadcast
      float xx1 = fmaxf(bi.x, bj.x);
      float yy1 = fmaxf(bi.y, bj.y);
      float xx2 = fminf(bi.z, bj.z);
      float yy2 = fminf(bi.w, bj.w);
      float inter = fmaxf(xx2 - xx1, 0.0f) * fmaxf(yy2 - yy1, 0.0f);
      float uni = C[r] - inter;           // area_i + area_j - inter (WMMA result)
      // iou > 0.5  <=>  inter > 0.5*max(uni,1e-9)   (divisor strictly positive)
      bool pred = (inter > NMS_THRESH * fmaxf(uni, 1e-9f)) && (gj > gi);
      unsigned bal = __builtin_amdgcn_ballot_w32(pred);
      if ((lane == 0) && (bal != 0u)) {   // mask is pre-zeroed; store only non-trivial rows
        unsigned short* m16 = (unsigned short*)s_mask;   // 32 halfwords per 512-bit row
        m16[(i0 + r)     * 32 + tj] = (unsigned short)(bal & 0xFFFFu);  // lanes 0-15  -> row i0+r
        m16[(i0 + r + 8) * 32 + tj] = (unsigned short)(bal >> 16);      // lanes 16-31 -> row i0+r+8
      }
    }
  }
  __syncthreads();

  // ---- greedy serial scan over the bit matrix: one wave, 16 lanes hold the 512-bit state ----
  if (t < 32) {
    unsigned removed = 0u;
    for (int i = 0; i < NN; ++i) {
      unsigned rw = __builtin_amdgcn_readlane(removed, i >> 5);
      if (((rw >> (i & 31)) & 1u) == 0u) {          // box i still alive -> it suppresses
        if (lane < 16) removed |= s_mask[i * 16 + lane];
      }
    }
    if (lane < 16) s_keep[lane] = ~removed;
  }
  __syncthreads();

  // ---- scatter: per-(b,n) max over classes via int-viewed float atomicMax (scores > 0) ----
  bool kept = (s_keep[t >> 5] >> (t & 31)) & 1u;
  if (kept) {
    float s = s_val[t];
    int n = s_idx[t];
    atomicMax((int*)(wsmax + (size_t)b * NN + n), __float_as_int(s));
  }
}

// ---------------- Kernel 3: finalize outputs ----------------
__global__ void finalize_kernel(const float* __restrict__ wsmax, float* __restrict__ out) {
  int i = blockIdx.x * blockDim.x + threadIdx.x;
  if (i >= NB * NN) return;
  float m = wsmax[i];
  out[i] = m;                                           // max_conf (8,512)
  out[NB * NN + i] = (m >= SCORE_THRESH) ? 1.0f : 0.0f; // keep     (8,512)
}

extern "C" void kernel_launch(void* const* d_in, const int* in_sizes, int n_in,
                              void* d_out, int out_size, void* d_ws, size_t ws_size,
                              hipStream_t stream) {
  const float* boxes  = (const float*)d_in[0];   // (8,512,80,4) f32
  const float* logits = (const float*)d_in[1];   // (8,512,81)   f32
  const float* scale  = (const float*)d_in[2];   // (8,2)        f32
  float* out = (float*)d_out;

  float* wsmax    = (float*)d_ws;                // 8*512 floats  (per-(b,n) running max)
  float* scores_t = wsmax + NB * NN;             // 8*80*512 floats (transposed class scores)

  prep_kernel<<<(NB * NN + 127) / 128, 128, 0, stream>>>(logits, wsmax, scores_t);
  nms_kernel<<<NB * NUM_CLASSES, 512, 0, stream>>>(boxes, scale, scores_t, wsmax);
  finalize_kernel<<<(NB * NN + 255) / 256, 256, 0, stream>>>(wsmax, out);
}